// DirectCXLEmbedding_25683904430111
// MI455X (gfx1250) — compile-verified
//
#include <hip/hip_runtime.h>
#include <stdint.h>

// ---------------------------------------------------------------------------
// DirectCXLEmbedding forward: out[b, l, :] = weight[indices[b, l], :]
//   indices: int32 [16384, 50]   weight: fp32 [1e6, 64]   out: fp32 [16384,50,64]
// Memory-bound (≈420 MB @ 23.3 TB/s ≈ 18 us floor). Implemented with the
// CDNA5 Tensor Data Mover in GATHER mode: one D# gathers 8 table rows
// (32-bit row indices, 256 B contiguous per row) into LDS; waves drain LDS to
// the contiguous output with non-temporal b128 stores so the 192 MB L2 stays
// reserved for the 256 MB table (≈32% of row fetches are statistical repeats).
//
// EXACT=true variant (launched for this shape: 819,200 % 32 == 0) has zero
// tail handling: per descriptor it is one s_load_b256 of 8 indices, SALU
// descriptor packing, and a tensor_load_to_lds; drains are branch-free.
// ---------------------------------------------------------------------------

typedef __attribute__((ext_vector_type(4))) unsigned int v4u;
typedef __attribute__((ext_vector_type(8))) int          v8i;
typedef __attribute__((ext_vector_type(4))) int          v4i;
typedef __attribute__((ext_vector_type(4))) float        v4f;

#define EMBED_DIM_F     64                          // floats per row
#define ROW_BYTES       (EMBED_DIM_F * 4)           // 256 B per row
#define ROWS_PER_DESC   8                           // 32-bit gather indices -> 8 rows/D#
#define DESC_PER_WAVE   4                           // 32 rows per wave
#define ROWS_PER_WAVE   (ROWS_PER_DESC * DESC_PER_WAVE)
#define DESC_F4         (ROWS_PER_DESC * ROW_BYTES / 16)   // 128 float4 per tile
#define WAVES_PER_BLOCK 8                           // 256 threads, 32 KB LDS

#if __has_builtin(__builtin_amdgcn_tensor_load_to_lds) && \
    __has_builtin(__builtin_amdgcn_s_wait_tensorcnt)
#define USE_TDM 1
#else
#define USE_TDM 0
#endif

#if USE_TDM
// Build D# and issue one TDM gather of 'vc' rows with row ids r[0..7] -> LDS.
__device__ __forceinline__ void tdm_issue_desc(const float* __restrict__ wgt,
                                               uint32_t lds_addr, int tableRows,
                                               int vc, const int r[ROWS_PER_DESC])
{
    uint64_t ga = (uint64_t)(uintptr_t)wgt;    // table base = tile start (x=0)

    // ---- D# group 0 (128b): count=1 | gather_index_size=32b | gather_mode=1
    v4u g0;
    g0[0] = 0xC0000001u;                       // [31]=gather, [30]=idx32, count=1
    g0[1] = lds_addr;                          // LDS byte address of tile
    g0[2] = (uint32_t)ga;                      // global_addr[31:0]
    g0[3] = ((uint32_t)(ga >> 32) & 0x01FFFFFFu) | 0x80000000u; // addr[56:32]|type=2

    // ---- D# group 1 (256b)
    v8i g1;
    g1[0] = (int)(2u << 16);                   // wg_mask=0 (no cluster), data_size=4B
    g1[1] = (int)((uint32_t)EMBED_DIM_F << 16);// abar_addr=0, tensor_dim0.lo16=64
    g1[2] = (int)(((uint32_t)tableRows & 0xFFFFu) << 16); // dim0.hi16=0, dim1.lo16
    g1[3] = (int)((((uint32_t)tableRows >> 16) & 0xFFFFu) // dim1.hi16
                  | ((uint32_t)EMBED_DIM_F << 16));       // tile_dim0 = 64
    g1[4] = vc;                                // tile_dim1 = #valid gather indices
    g1[5] = EMBED_DIM_F;                       // tensor_dim0_stride = 64 elements
    g1[6] = 0;                                 // stride0.hi, stride1 (ignored: gather)
    g1[7] = 0;

    // ---- D# groups 2/3: 32-bit row indices 0..7
    v4i g2 = { r[0], r[1], r[2], r[3] };
    v4i g3 = { r[4], r[5], r[6], r[7] };

    // ---- extra SGPR group (toolchain 6-arg form): unused -> zeros
    v8i gx = { 0, 0, 0, 0, 0, 0, 0, 0 };

    __builtin_amdgcn_tensor_load_to_lds(g0, g1, g2, g3, gx, 0);
}

// Fetch 8 gather indices (wave-uniform) and issue the descriptor.
template<bool EXACT>
__device__ __forceinline__ void tdm_issue_gather(const float* __restrict__ wgt,
                                                 uint32_t lds_addr, int tableRows,
                                                 const int* __restrict__ idx,
                                                 int rowBase, int totalRows)
{
    const int rb  = __builtin_amdgcn_readfirstlane(rowBase);
    const int* ip = idx + rb;
    int r[ROWS_PER_DESC];

    if (EXACT || (totalRows - rb) >= ROWS_PER_DESC) {
        // Full tile: 8 unconditional consecutive uniform loads (-> s_load_b256).
#pragma unroll
        for (int k = 0; k < ROWS_PER_DESC; ++k)
            r[k] = __builtin_amdgcn_readfirstlane(ip[k]);
        tdm_issue_desc(wgt, lds_addr, tableRows, ROWS_PER_DESC, r);
    } else if ((totalRows - rb) > 0) {
        // Uniform tail: clamp offsets with scalar selects.
        const int vc = totalRows - rb;
#pragma unroll
        for (int k = 0; k < ROWS_PER_DESC; ++k) {
            int off = (k < vc) ? k : 0;
            r[k] = __builtin_amdgcn_readfirstlane(ip[off]);
        }
        tdm_issue_desc(wgt, lds_addr, tableRows, vc, r);
    }
    // rem <= 0: trailing wave, nothing to fetch.
}

// Drain one 2 KB LDS tile to the contiguous output region.
// Chunk layout: 4 passes x (32 lanes x 16 B) = 512 B fully-coalesced stores.
template<bool EXACT>
__device__ __forceinline__ void stream_out(const v4f* __restrict__ buf,
                                           float* __restrict__ out,
                                           int rowBase, int totalRows, int lane)
{
    const int rb  = __builtin_amdgcn_readfirstlane(rowBase);
    const int rem = EXACT ? ROWS_PER_DESC : (totalRows - rb);
    v4f* dst = (v4f*)(out + (long)rb * EMBED_DIM_F);

    if (EXACT || rem >= ROWS_PER_DESC) {       // branch-free full tile
#pragma unroll
        for (int c = 0; c < DESC_F4 / 32; ++c) {
            int f4 = c * 32 + lane;
            __builtin_nontemporal_store(buf[f4], dst + f4);   // NT: keep L2 for table
        }
    } else if (rem > 0) {                      // uniform tail path
        int nf4 = rem * (ROW_BYTES / 16);
#pragma unroll
        for (int c = 0; c < DESC_F4 / 32; ++c) {
            int f4 = c * 32 + lane;
            if (f4 < nf4)
                __builtin_nontemporal_store(buf[f4], dst + f4);
        }
    }
}

template<bool EXACT>
__global__ __launch_bounds__(WAVES_PER_BLOCK * 32, 1)
void emb_gather_tdm(const int* __restrict__ idx, const float* __restrict__ wgt,
                    float* __restrict__ out, int totalRows, int tableRows)
{
    __shared__ v4f smem[WAVES_PER_BLOCK * 2 * DESC_F4];   // 2x2KB per wave
    const int lane = threadIdx.x & 31;
    const int wid  = threadIdx.x >> 5;
    const int gw   = blockIdx.x * WAVES_PER_BLOCK + wid;
    const int base = gw * ROWS_PER_WAVE;       // max 819,200: fits int

    v4f* buf0 = smem + wid * 2 * DESC_F4;
    v4f* buf1 = buf0 + DESC_F4;
    uint32_t lds0 = (uint32_t)(uintptr_t)buf0;   // flat LDS addr truncates to offset
    uint32_t lds1 = (uint32_t)(uintptr_t)buf1;

    // Software-pipelined double buffer. TENSORcnt completes in order per wave,
    // so s_wait_tensorcnt(1) == "oldest gather landed in LDS".
    tdm_issue_gather<EXACT>(wgt, lds0, tableRows, idx, base,                     totalRows);
    tdm_issue_gather<EXACT>(wgt, lds1, tableRows, idx, base + 1 * ROWS_PER_DESC, totalRows);
    __builtin_amdgcn_s_wait_tensorcnt(1);
    stream_out<EXACT>(buf0, out, base,                     totalRows, lane);

    tdm_issue_gather<EXACT>(wgt, lds0, tableRows, idx, base + 2 * ROWS_PER_DESC, totalRows);
    __builtin_amdgcn_s_wait_tensorcnt(1);
    stream_out<EXACT>(buf1, out, base + 1 * ROWS_PER_DESC, totalRows, lane);

    tdm_issue_gather<EXACT>(wgt, lds1, tableRows, idx, base + 3 * ROWS_PER_DESC, totalRows);
    __builtin_amdgcn_s_wait_tensorcnt(1);
    stream_out<EXACT>(buf0, out, base + 2 * ROWS_PER_DESC, totalRows, lane);

    __builtin_amdgcn_s_wait_tensorcnt(0);
    stream_out<EXACT>(buf1, out, base + 3 * ROWS_PER_DESC, totalRows, lane);
}
#endif  // USE_TDM

// Fallback / portability path: plain b128 gather, one float4 per thread.
__global__ __launch_bounds__(256)
void emb_gather_vmem(const int* __restrict__ idx, const float* __restrict__ wgt,
                     float* __restrict__ out, long totalRows)
{
    long g = (long)blockIdx.x * blockDim.x + threadIdx.x;   // float4 granularity
    long totalF4 = totalRows * (EMBED_DIM_F / 4);
    if (g >= totalF4) return;
    long row  = g >> 4;                 // 16 float4 per 64-float row
    int  part = (int)(g & 15);
    int  e    = idx[row];
    const v4f* src = (const v4f*)(wgt + (long)e * EMBED_DIM_F);
    v4f v = src[part];
    __builtin_nontemporal_store(v, ((v4f*)out) + g);
}

extern "C" void kernel_launch(void* const* d_in, const int* in_sizes, int n_in,
                              void* d_out, int out_size, void* d_ws, size_t ws_size,
                              hipStream_t stream)
{
    const int*   idx = (const int*)d_in[0];
    const float* wgt = (const float*)d_in[1];
    float*       out = (float*)d_out;

    const int totalRows = in_sizes[0];                 // 16384*50 = 819,200
    const int tableRows = in_sizes[1] / EMBED_DIM_F;   // 1,000,000

#if USE_TDM
    long waves  = ((long)totalRows + ROWS_PER_WAVE - 1) / ROWS_PER_WAVE;  // 25,600
    int  blocks = (int)((waves + WAVES_PER_BLOCK - 1) / WAVES_PER_BLOCK); // 3,200
    if (totalRows % ROWS_PER_WAVE == 0)                // this shape: always true
        emb_gather_tdm<true><<<blocks, WAVES_PER_BLOCK * 32, 0, stream>>>(
            idx, wgt, out, totalRows, tableRows);
    else
        emb_gather_tdm<false><<<blocks, WAVES_PER_BLOCK * 32, 0, stream>>>(
            idx, wgt, out, totalRows, tableRows);
#else
    long totalF4 = (long)totalRows * (EMBED_DIM_F / 4);
    int  blocks  = (int)((totalF4 + 255) / 256);
    emb_gather_vmem<<<blocks, 256, 0, stream>>>(idx, wgt, out, (long)totalRows);
#endif
    (void)n_in; (void)out_size; (void)d_ws; (void)ws_size; (void)tableRows;
}